// GCN_80238579024176
// MI455X (gfx1250) — compile-verified
//
#include <hip/hip_runtime.h>
#include <hip/hip_bf16.h>

#define N_NODES   100000
#define N_EDGES   3200000
#define N_GRAPHS  64
#define IN_DIM    8
#define HIDDEN    128
#define N_CLASSES 10

typedef __attribute__((ext_vector_type(2))) float v2f;
typedef __attribute__((ext_vector_type(8))) float v8f;

// ---------------------------------------------------------------------------
// Degree / normalization precompute
// ---------------------------------------------------------------------------
__global__ void k_fill(float* __restrict__ p, long long n, float v) {
    long long i = (long long)blockIdx.x * blockDim.x + threadIdx.x;
    if (i < n) p[i] = v;
}

__global__ void k_deg_accum(const int* __restrict__ dst, float* __restrict__ deg) {
    long long e = (long long)blockIdx.x * blockDim.x + threadIdx.x;
    if (e < N_EDGES) atomicAdd(&deg[dst[e]], 1.0f);
}

// deg (in-place) -> dis = rsqrt(deg); self_coef = 1/deg
__global__ void k_deg_finalize(float* __restrict__ deg_dis, float* __restrict__ self_coef) {
    long long n = (long long)blockIdx.x * blockDim.x + threadIdx.x;
    if (n < N_NODES) {
        float d = deg_dis[n];
        deg_dis[n]   = rsqrtf(d);
        self_coef[n] = 1.0f / d;
    }
}

__global__ void k_enorm(const int* __restrict__ src, const int* __restrict__ dst,
                        const float* __restrict__ dis, float* __restrict__ enorm) {
    long long e = (long long)blockIdx.x * blockDim.x + threadIdx.x;
    if (e < N_EDGES) enorm[e] = dis[src[e]] * dis[dst[e]];
}

// ---------------------------------------------------------------------------
// Dense GEMM: Out[N_NODES x 128] = A[N_NODES x K] * W[K x 128]
// via V_WMMA_F32_16X16X4_F32 (fp32 matrix pipe, exact fp32 semantics).
// Block: 256 thr (8 waves) = one 16-row node slab; wave w -> 16-col tile.
// A slab (16 x K, contiguous in memory) staged in LDS once per block.
// ---------------------------------------------------------------------------
__global__ __launch_bounds__(256) void k_gemm_wmma(const float* __restrict__ A,
                                                   const float* __restrict__ W,
                                                   float* __restrict__ Out,
                                                   int K) {
    __shared__ float lda[16 * HIDDEN];             // up to K = 128 -> 8 KB
    const int m0   = blockIdx.x * 16;
    const int tid  = threadIdx.x;
    const int lane = tid & 31;
    const int n0   = (tid >> 5) * 16;              // wave id * 16

    // 16 rows of A are one contiguous 16*K block (row-major, ld == K)
    const float* Ab = A + (size_t)m0 * K;
    for (int i = tid; i < 16 * K; i += 256) lda[i] = Ab[i];
    __syncthreads();

    const int half = lane >> 4;                    // 0: K{0,1}, 1: K{2,3}
    const int ln   = lane & 15;

    v8f c = {};
    for (int k0 = 0; k0 < K; k0 += 4) {
        const int kk = k0 + 2 * half;
        v2f a, b;
        // A 16x4 layout: lane(0-15)=row, vgpr pair = consecutive K
        a.x = lda[ln * K + kk];
        a.y = lda[ln * K + kk + 1];
        // B 4x16 layout: lane(0-15)=col, vgpr pair = consecutive K
        b.x = W[(size_t)kk * HIDDEN + n0 + ln];
        b.y = W[(size_t)(kk + 1) * HIDDEN + n0 + ln];
        c = __builtin_amdgcn_wmma_f32_16x16x4_f32(false, a, false, b,
                                                  (short)0, c, false, false);
    }

    // C/D layout: vgpr r -> row r (lanes 0-15) / row r+8 (lanes 16-31)
    float* Ob = Out + (size_t)m0 * HIDDEN + n0 + ln;
#pragma unroll
    for (int r = 0; r < 8; ++r)
        Ob[(size_t)(r + 8 * half) * HIDDEN] = c[r];
}

// ---------------------------------------------------------------------------
// acc = self_coef[n] * hLin + bias  (seeds the edge accumulator)
// ---------------------------------------------------------------------------
__global__ void k_preadd(const float* __restrict__ hLin, const float* __restrict__ self_coef,
                         const float* __restrict__ bias, float* __restrict__ acc) {
    long long idx = (long long)blockIdx.x * blockDim.x + threadIdx.x;
    if (idx >= (long long)N_NODES * HIDDEN) return;
    int n = (int)(idx >> 7);
    int t = (int)(idx & 127);
    acc[idx] = self_coef[n] * hLin[idx] + bias[t];
}

// ---------------------------------------------------------------------------
// Edge scatter: one wave per edge; lane reads float4 (full 512B row touched
// by the wave), 4 coalesced f32 atomics into acc[dst]. h is L2-resident.
// ---------------------------------------------------------------------------
__global__ __launch_bounds__(256) void k_edge_scatter(const float* __restrict__ hLin,
                                                      const float* __restrict__ enorm,
                                                      const int* __restrict__ src,
                                                      const int* __restrict__ dst,
                                                      float* __restrict__ acc) {
    const int lane = threadIdx.x & 31;
    const long long e = (long long)blockIdx.x * 8 + (threadIdx.x >> 5);
    if (e >= N_EDGES) return;
    const int   s = src[e];
    const int   d = dst[e];
    const float w = enorm[e];
    const float4 v = reinterpret_cast<const float4*>(hLin + (size_t)s * HIDDEN)[lane];
    float* o = acc + (size_t)d * HIDDEN + lane * 4;
    atomicAdd(o + 0, v.x * w);
    atomicAdd(o + 1, v.y * w);
    atomicAdd(o + 2, v.z * w);
    atomicAdd(o + 3, v.w * w);
}

__global__ void k_relu(float* __restrict__ h, long long n) {
    long long i = (long long)blockIdx.x * blockDim.x + threadIdx.x;
    if (i < n) h[i] = fmaxf(h[i], 0.0f);
}

// ---------------------------------------------------------------------------
// Mean pool: batch is sorted -> each graph-block binary-searches its node
// range and reduces contention-free (no atomics).
// ---------------------------------------------------------------------------
__global__ __launch_bounds__(128) void k_pool(const float* __restrict__ h,
                                              const int* __restrict__ batch,
                                              float* __restrict__ pooled) {
    const int g = blockIdx.x;
    const int t = threadIdx.x;
    int lo = 0, hi = N_NODES;
    while (lo < hi) { int m = (lo + hi) >> 1; if (batch[m] < g) lo = m + 1; else hi = m; }
    const int start = lo;
    hi = N_NODES;
    while (lo < hi) { int m = (lo + hi) >> 1; if (batch[m] < g + 1) lo = m + 1; else hi = m; }
    const int end = lo;

    float acc = 0.0f;
    for (int n = start; n < end; ++n) acc += h[(size_t)n * HIDDEN + t];
    pooled[(size_t)g * HIDDEN + t] = acc / fmaxf((float)(end - start), 1.0f);
}

// ---------------------------------------------------------------------------
// Head: logits = pooled @ Wout + bout, then stable log_softmax. One wave/graph.
// ---------------------------------------------------------------------------
__global__ __launch_bounds__(32) void k_head(const float* __restrict__ pooled,
                                             const float* __restrict__ Wout,
                                             const float* __restrict__ bout,
                                             float* __restrict__ out) {
    __shared__ float l[N_CLASSES];
    const int g = blockIdx.x;
    const int c = threadIdx.x;
    if (c < N_CLASSES) {
        float acc = bout[c];
        const float* p = pooled + (size_t)g * HIDDEN;
        for (int k = 0; k < HIDDEN; ++k) acc += p[k] * Wout[k * N_CLASSES + c];
        l[c] = acc;
    }
    __syncthreads();
    if (c < N_CLASSES) {
        float m = l[0];
        for (int i = 1; i < N_CLASSES; ++i) m = fmaxf(m, l[i]);
        float s = 0.0f;
        for (int i = 0; i < N_CLASSES; ++i) s += expf(l[i] - m);
        out[g * N_CLASSES + c] = l[c] - (m + logf(s));
    }
}

// ---------------------------------------------------------------------------
// Host orchestration
// ---------------------------------------------------------------------------
extern "C" void kernel_launch(void* const* d_in, const int* in_sizes, int n_in,
                              void* d_out, int out_size, void* d_ws, size_t ws_size,
                              hipStream_t stream) {
    (void)in_sizes; (void)n_in; (void)out_size; (void)ws_size;

    const float* x     = (const float*)d_in[0];
    const int*   edges = (const int*)d_in[1];
    const int*   src   = edges;
    const int*   dst   = edges + N_EDGES;
    const int*   batch = (const int*)d_in[2];
    const float* W[5]  = {(const float*)d_in[3], (const float*)d_in[5], (const float*)d_in[7],
                          (const float*)d_in[9], (const float*)d_in[11]};
    const float* b[5]  = {(const float*)d_in[4], (const float*)d_in[6], (const float*)d_in[8],
                          (const float*)d_in[10], (const float*)d_in[12]};
    const float* Wout  = (const float*)d_in[13];
    const float* bout  = (const float*)d_in[14];
    float*       out   = (float*)d_out;

    // workspace layout (floats)
    float* ws      = (float*)d_ws;
    float* deg_dis = ws;                                   // deg, then dis
    float* selfc   = deg_dis + N_NODES;
    float* enorm   = selfc + N_NODES;
    float* hCur    = enorm + N_EDGES;                      // N*128
    float* hLin    = hCur + (size_t)N_NODES * HIDDEN;      // N*128
    float* pooled  = hLin + (size_t)N_NODES * HIDDEN;      // 64*128

    const long long NH = (long long)N_NODES * HIDDEN;

    // normalization precompute
    k_fill<<<(N_NODES + 255) / 256, 256, 0, stream>>>(deg_dis, N_NODES, 1.0f);
    k_deg_accum<<<(N_EDGES + 255) / 256, 256, 0, stream>>>(dst, deg_dis);
    k_deg_finalize<<<(N_NODES + 255) / 256, 256, 0, stream>>>(deg_dis, selfc);
    k_enorm<<<(N_EDGES + 255) / 256, 256, 0, stream>>>(src, dst, deg_dis, enorm);

    // 5 GCN layers
    const float* in = x;
    int K = IN_DIM;
    for (int L = 0; L < 5; ++L) {
        k_gemm_wmma<<<N_NODES / 16, 256, 0, stream>>>(in, W[L], hLin, K);
        k_preadd<<<(int)((NH + 255) / 256), 256, 0, stream>>>(hLin, selfc, b[L], hCur);
        k_edge_scatter<<<N_EDGES / 8, 256, 0, stream>>>(hLin, enorm, src, dst, hCur);
        k_relu<<<(int)((NH + 255) / 256), 256, 0, stream>>>(hCur, NH);
        in = hCur;
        K = HIDDEN;
    }

    // pool + classifier head
    k_pool<<<N_GRAPHS, 128, 0, stream>>>(hCur, batch, pooled);
    k_head<<<N_GRAPHS, 32, 0, stream>>>(pooled, Wout, bout, out);
}